// Mixture_67688684585185
// MI455X (gfx1250) — compile-verified
//
#include <hip/hip_runtime.h>
#include <hip/hip_bf16.h>

#define NPART  128
#define NDIM   128   // NSTART == NEND == 128
#define BATCH  4096

typedef __attribute__((ext_vector_type(2))) float v2f;
typedef __attribute__((ext_vector_type(8))) float v8f;

// ---------------------------------------------------------------------------
// Kernel 1: softmax over logp[128] -> p_soft[128]  (one block, 128 threads)
// ---------------------------------------------------------------------------
__global__ void softmax128_kernel(const float* __restrict__ logp,
                                  float* __restrict__ p_soft) {
    __shared__ float sh[NPART];
    const int t = threadIdx.x;
    float v = logp[t];            // logp is [128,1]
    sh[t] = v;
    __syncthreads();
    for (int s = 64; s > 0; s >>= 1) {
        if (t < s) sh[t] = fmaxf(sh[t], sh[t + s]);
        __syncthreads();
    }
    const float m = sh[0];
    __syncthreads();
    const float e = expf(v - m);
    sh[t] = e;
    __syncthreads();
    for (int s = 64; s > 0; s >>= 1) {
        if (t < s) sh[t] += sh[t + s];
        __syncthreads();
    }
    p_soft[t] = e / sh[0];
}

// ---------------------------------------------------------------------------
// Kernel 2: bucket samples by particle (counting sort).
//   offsets[p] .. offsets[p+1] delimit order[] entries whose randind == p.
//   One block of 128 threads; counts + cursors live in LDS.
// ---------------------------------------------------------------------------
__global__ void bucket_kernel(const int* __restrict__ randind,
                              int* __restrict__ offsets,
                              int* __restrict__ order) {
    __shared__ int cnt[NPART];
    __shared__ int off[NPART + 1];
    const int t = threadIdx.x;
    cnt[t] = 0;
    __syncthreads();
    for (int b = t; b < BATCH; b += NPART)
        atomicAdd(&cnt[randind[b]], 1);
    __syncthreads();
    if (t == 0) {                 // serial exclusive scan over 128 entries: trivial
        int acc = 0;
        for (int i = 0; i < NPART; ++i) { off[i] = acc; acc += cnt[i]; }
        off[NPART] = acc;
    }
    __syncthreads();
    cnt[t] = off[t];              // reuse cnt[] as per-particle cursor
    __syncthreads();
    for (int b = t; b < BATCH; b += NPART) {
        const int p = randind[b];
        const int slot = atomicAdd(&cnt[p], 1);
        order[slot] = b;
    }
    offsets[t] = off[t];
    if (t == 0) offsets[NPART] = off[NPART];
}

// ---------------------------------------------------------------------------
// Kernel 3: weights[b] = softmax(logp)[randind[b]]; pass randind through.
// ---------------------------------------------------------------------------
__global__ void weights_kernel(const float* __restrict__ p_soft,
                               const int* __restrict__ randind,
                               float* __restrict__ w_out,
                               int* __restrict__ r_out) {
    const int i = blockIdx.x * blockDim.x + threadIdx.x;
    if (i < BATCH) {
        const int r = randind[i];
        w_out[i] = p_soft[r];
        r_out[i] = r;
    }
}

// ---------------------------------------------------------------------------
// Kernel 4: per-particle GEMM with V_WMMA_F32_16X16X4_F32.
//   For particle p and its bucket {b_m}:
//     out[b_m, n] = pos[p, n] + sum_j gaussian[b_m, j] * sigma[p, n, j]
//   Block = 256 threads = 8 wave32; wave w owns N-tile w (16 columns),
//   loops over 16-row M-tiles of the bucket. K=128 in 32 steps of 4.
//
//   f32 WMMA fragment layout (ISA 7.12.2):
//     A (16x4): lane L -> row M = L%16, K in {2*(L/16), 2*(L/16)+1}
//     B (4x16): lane L -> col N = L%16, same K pair  -> B col n == sigma row n
//     C/D     : VGPR v -> M = v + 8*(L/16), N = L%16
// ---------------------------------------------------------------------------
__global__ void gemm_wmma_kernel(const float* __restrict__ Y,      // [BATCH][128]
                                 const float* __restrict__ pos,    // [128][128]
                                 const float* __restrict__ sigma,  // [128][128][128]
                                 const int* __restrict__ offsets,  // [129]
                                 const int* __restrict__ order,    // [BATCH]
                                 float* __restrict__ out) {        // [BATCH][128]
    const int p     = blockIdx.x;
    const int start = offsets[p];
    const int cnt   = offsets[p + 1] - start;
    if (cnt == 0) return;                       // uniform per block

    const int wave  = threadIdx.x >> 5;         // 0..7 -> N-tile
    const int lane  = threadIdx.x & 31;
    const int lhalf = lane >> 4;                // selects K pair
    const int lmod  = lane & 15;

    const int ncol = wave * 16 + lmod;          // this lane's output column
    const float* __restrict__ sigRow = sigma + ((size_t)p * NDIM + ncol) * NDIM;
    const float  cpos = pos[p * NDIM + ncol];

    const int mtiles = (cnt + 15) >> 4;
    for (int mt = 0; mt < mtiles; ++mt) {
        const int midx = mt * 16 + lmod;
        // padded rows load a safe row; their stores are masked below
        const int bA = (midx < cnt) ? order[start + midx] : order[start];
        const float* __restrict__ yRow = Y + (size_t)bA * NDIM;

        v8f acc;
#pragma unroll
        for (int v = 0; v < 8; ++v) acc[v] = cpos;   // C = pos broadcast over M

#pragma unroll
        for (int k = 0; k < NDIM; k += 4) {
            const int kk = k + 2 * lhalf;
            const v2f a = *(const v2f*)(yRow + kk);
            const v2f b = *(const v2f*)(sigRow + kk);
            acc = __builtin_amdgcn_wmma_f32_16x16x4_f32(
                /*neg_a=*/false, a, /*neg_b=*/false, b,
                /*c_mod=*/(short)0, acc, /*reuse_a=*/false, /*reuse_b=*/false);
        }

#pragma unroll
        for (int v = 0; v < 8; ++v) {
            const int mg = mt * 16 + v + 8 * lhalf;
            if (mg < cnt) {
                const int b = order[start + mg];
                out[(size_t)b * NDIM + ncol] = acc[v];
            }
        }
    }
}

// ---------------------------------------------------------------------------
// Host-side launcher.
// Inputs (setup_inputs order): 0=gaussian f32[4096*128], 1=logp f32[128],
//   2=pos f32[128*128], 3=sigma f32[128*128*128], 4=randind i32[4096]
// Output tuple flat: out f32[4096*128] | weights f32[4096] | randind i32[4096]
// ---------------------------------------------------------------------------
extern "C" void kernel_launch(void* const* d_in, const int* in_sizes, int n_in,
                              void* d_out, int out_size, void* d_ws, size_t ws_size,
                              hipStream_t stream) {
    const float* gaussian = (const float*)d_in[0];
    const float* logp     = (const float*)d_in[1];
    const float* pos      = (const float*)d_in[2];
    const float* sigma    = (const float*)d_in[3];
    const int*   randind  = (const int*)d_in[4];

    float* out_main = (float*)d_out;                               // [4096*128]
    float* out_w    = out_main + (size_t)BATCH * NDIM;             // [4096]
    int*   out_r    = (int*)(out_w + BATCH);                       // [4096] (int32 bits)

    // Workspace layout
    float* p_soft  = (float*)d_ws;                                 // 128 floats
    int*   offsets = (int*)((char*)d_ws + 512);                    // 129 ints
    int*   order   = (int*)((char*)d_ws + 2048);                   // 4096 ints

    softmax128_kernel<<<1, NPART, 0, stream>>>(logp, p_soft);
    bucket_kernel<<<1, NPART, 0, stream>>>(randind, offsets, order);
    weights_kernel<<<(BATCH + 255) / 256, 256, 0, stream>>>(p_soft, randind, out_w, out_r);
    gemm_wmma_kernel<<<NPART, 256, 0, stream>>>(gaussian, pos, sigma, offsets, order, out_main);
}